// Attention_32401233281111
// MI455X (gfx1250) — compile-verified
//
#include <hip/hip_runtime.h>
#include <math.h>

#define IN_SIZE 512
#define HIDDEN  16
#define NSEQ    64
#define ZSTRIDE 516   // 512 + 4 dword pad: A-tile reads & pooling reads hit distinct banks

typedef __attribute__((ext_vector_type(2))) float v2f;
typedef __attribute__((ext_vector_type(8))) float v8f;

// 16-byte payload type matching the async-to-LDS builtin's parameter:
//   '__attribute__((__vector_size__(4 * sizeof(int)))) int __device__ *'
typedef int b128i __attribute__((vector_size(16)));
typedef __attribute__((address_space(1))) b128i* gptr_b128;   // global (AS1)
typedef __attribute__((address_space(3))) b128i* lptr_b128;   // LDS (AS3)

#if __has_builtin(__builtin_amdgcn_global_load_async_to_lds_b128) && \
    __has_builtin(__builtin_amdgcn_s_wait_asynccnt)
#define USE_ASYNC_COPY 1
#else
#define USE_ASYNC_COPY 0
#endif

extern "C" __global__ __launch_bounds__(256)
void attn_pool_kernel(const float* __restrict__ z,
                      const float* __restrict__ W1,
                      const float* __restrict__ b1,
                      const float* __restrict__ W2,
                      float* __restrict__ pooled,    // [B, 512]
                      float* __restrict__ beta_out)  // [B, 64]
{
    extern __shared__ float lds[];
    float* lds_z = lds;                           // 64 * ZSTRIDE  (~129 KB)
    float* lds_w = lds + NSEQ * ZSTRIDE;          // 64 scores
    float* lds_b = lds_w + NSEQ;                  // 64 betas

    const int b    = blockIdx.x;
    const int tid  = threadIdx.x;
    const int lane = tid & 31;
    const int half = lane >> 4;                   // which 16-lane half of the wave
    const int l16  = lane & 15;
    const int wave = __builtin_amdgcn_readfirstlane(tid >> 5);

    // ---- Stage 1: stream z[b] (64x512 f32 = 128KB) into LDS, once per block ----
#if USE_ASYNC_COPY
    {
        gptr_b128 zg = (gptr_b128)(z + (size_t)b * NSEQ * IN_SIZE);
        #pragma unroll 4
        for (int i = 0; i < 32; ++i) {            // 8192 x b128 / 256 threads
            int idx = i * 256 + tid;
            int row = idx >> 7;                   // 128 float4 per row
            int c4  = idx & 127;
            __builtin_amdgcn_global_load_async_to_lds_b128(
                zg + idx,
                (lptr_b128)&lds_z[row * ZSTRIDE + c4 * 4],
                0, 0);
        }
        __builtin_amdgcn_s_wait_asynccnt(0);
    }
#else
    {
        const float4* zg = (const float4*)(z + (size_t)b * NSEQ * IN_SIZE);
        #pragma unroll 4
        for (int i = 0; i < 32; ++i) {
            int idx = i * 256 + tid;
            float4 v = zg[idx];
            int row = idx >> 7;
            int c4  = idx & 127;
            *(float4*)&lds_z[row * ZSTRIDE + c4 * 4] = v;
        }
    }
#endif
    __syncthreads();

    // ---- Stage 2: C = z_tile @ W1 via v_wmma_f32_16x16x4_f32, waves 0..3 ----
    // W1 (32 KB) is shared by all 2048 blocks -> persistent in L2 / WGP$; read it
    // straight from global so LDS stays under 160 KB and 2 workgroups fit per WGP.
    if (wave < 4) {
        const int row = wave * 16 + l16;          // A-matrix row for this lane
        v8f acc = {};
        for (int k = 0; k < IN_SIZE; k += 4) {
            const int kk = k + 2 * half;          // lanes 0-15: K=k,k+1; 16-31: K=k+2,k+3
            v2f a = *(const v2f*)&lds_z[row * ZSTRIDE + kk];
            v2f bm;
            bm.x = W1[kk * HIDDEN + l16];         // B 4x16: row K, col N=l16 (L0-cached)
            bm.y = W1[(kk + 1) * HIDDEN + l16];
            acc = __builtin_amdgcn_wmma_f32_16x16x4_f32(
                false, a, false, bm, (short)0, acc, false, false);
        }
        // acc[r] = C[m = r + 8*half][n = l16];  h = tanh(C + b1);  score = h . W2
        const float bias = b1[l16];
        const float w2v  = W2[l16];
        #pragma unroll
        for (int r = 0; r < 8; ++r) {
            float h = tanhf(acc[r] + bias);
            float t = h * w2v;
            t += __shfl_xor(t, 1, 32);            // reduce across n within 16-lane halves
            t += __shfl_xor(t, 2, 32);
            t += __shfl_xor(t, 4, 32);
            t += __shfl_xor(t, 8, 32);
            if (l16 == 0)
                lds_w[wave * 16 + half * 8 + r] = t;
        }
    }
    __syncthreads();

    // ---- Stage 3: softmax over N=64 (redundant per-thread scan, broadcast reads) ----
    float mx = -3.0e38f;
    for (int j = 0; j < NSEQ; ++j) mx = fmaxf(mx, lds_w[j]);
    float s = 0.0f;
    for (int j = 0; j < NSEQ; ++j) s += __expf(lds_w[j] - mx);
    const float inv = 1.0f / s;
    if (tid < NSEQ) {
        float bv = __expf(lds_w[tid] - mx) * inv;
        lds_b[tid] = bv;
        __builtin_nontemporal_store(bv, &beta_out[(size_t)b * NSEQ + tid]);
    }
    __syncthreads();

    // ---- Stage 4: pooled[d] = sum_n beta[n] * z[n][d]; 2 columns per thread ----
    {
        const int d = tid * 2;
        v2f acc2 = {0.0f, 0.0f};
        for (int n = 0; n < NSEQ; ++n) {
            float bn = lds_b[n];
            v2f  zz = *(const v2f*)&lds_z[n * ZSTRIDE + d];
            acc2.x += bn * zz.x;
            acc2.y += bn * zz.y;
        }
        __builtin_nontemporal_store(acc2, (v2f*)&pooled[(size_t)b * IN_SIZE + d]);
    }
}

extern "C" void kernel_launch(void* const* d_in, const int* in_sizes, int n_in,
                              void* d_out, int out_size, void* d_ws, size_t ws_size,
                              hipStream_t stream) {
    const float* z  = (const float*)d_in[0];
    const float* W1 = (const float*)d_in[1];
    const float* b1 = (const float*)d_in[2];
    const float* W2 = (const float*)d_in[3];
    const int B = in_sizes[0] / (NSEQ * IN_SIZE);   // 2048

    float* pooled = (float*)d_out;                           // [B,512]
    float* beta   = (float*)d_out + (size_t)B * IN_SIZE;     // [B,64]

    const size_t shmem =
        (size_t)(NSEQ * ZSTRIDE + NSEQ + NSEQ) * sizeof(float);  // ~129.6 KB -> 2 WG/WGP

    attn_pool_kernel<<<B, 256, shmem, stream>>>(z, W1, b1, W2, pooled, beta);
}